// AttentionLayer_78829829751023
// MI455X (gfx1250) — compile-verified
//
#include <hip/hip_runtime.h>
#include <math.h>

#define B_    64
#define S_    4096
#define ENC_  512
#define DEC_  512
#define CHUNKS 8            // blocks per batch in the fused pass
#define WAVES  8            // waves per block (256 threads, wave32)
#define ROWS_PER_WAVE (S_ / CHUNKS / WAVES)   // 64
#define REC_STRIDE 520      // floats per partial record (2 + 512, padded)

typedef __attribute__((ext_vector_type(2))) float v2f;
typedef __attribute__((ext_vector_type(8))) float v8f;

// ---------------------------------------------------------------------------
// Kernel 1: sub[b,e] = sum_d dec[b,d] * W[e,d]   (GEMM 64x512x512, fp32 WMMA)
// One wave per 16x16 output tile. A = dec tile (16x4), B[d,e] = W[e,d] (4x16).
// A/B layout: VGPR v, lanes 0-15 hold K = v; lanes 16-31 hold K = v + 2.
// D layout:   VGPR v, lanes 0-15 hold M = v, N = lane; lanes 16-31 M = v + 8.
// ---------------------------------------------------------------------------
__global__ __launch_bounds__(32) void sub_gemm_wmma(const float* __restrict__ dec,
                                                    const float* __restrict__ W,
                                                    float* __restrict__ sub) {
    const int b0   = blockIdx.x * 16;
    const int e0   = blockIdx.y * 16;
    const int lane = threadIdx.x;        // 0..31, EXEC all ones
    const int lm   = lane & 15;
    const int hs2  = (lane >> 4) * 2;    // 0 or 2

    const float* decRow = dec + (size_t)(b0 + lm) * DEC_;  // A row (M = lm)
    const float* wRow   = W   + (size_t)(e0 + lm) * DEC_;  // B col (N = lm), W row-major

    v8f c = {};
    for (int k = 0; k < DEC_; k += 4) {
        v2f a, b;
        a[0] = decRow[k + hs2 + 0];      // K = k + 0 + 2*half
        a[1] = decRow[k + hs2 + 1];      // K = k + 1 + 2*half
        b[0] = wRow[k + hs2 + 0];        // B[K][e0+lm] = W[e0+lm][K]
        b[1] = wRow[k + hs2 + 1];
        c = __builtin_amdgcn_wmma_f32_16x16x4_f32(false, a, false, b,
                                                  (short)0, c, false, false);
    }
#pragma unroll
    for (int v = 0; v < 8; ++v) {
        const int row = b0 + v + 8 * (lane >> 4);          // M
        sub[(size_t)row * ENC_ + e0 + lm] = c[v];          // N = lm
    }
}

// ---------------------------------------------------------------------------
// Kernel 2: fused scores + online softmax + weighted-sum partials.
// Single pass over encoderFeature (the 512 MiB tensor). Each wave owns a set
// of rows; lane l owns elements e = 4l + 128k (k=0..3) of each 512-row.
// Raw scores go to the attn output region (normalized later by kernel 4).
// ---------------------------------------------------------------------------
__global__ __launch_bounds__(256) void attn_pass1(const float* __restrict__ enc,
                                                  const float* __restrict__ sub,
                                                  float* __restrict__ attn_raw,
                                                  float* __restrict__ recs) {
    const int b     = blockIdx.x;
    const int chunk = blockIdx.y;
    const int tid   = threadIdx.x;
    const int w     = tid >> 5;
    const int lane  = tid & 31;
    const int rowBase = chunk * (WAVES * ROWS_PER_WAVE);

    __shared__ float smax[WAVES];
    __shared__ float sden[WAVES];
    __shared__ float sacc[WAVES][ENC_];

    // sub[b] resident in registers: 16 floats/lane, same pattern as enc rows.
    const float4* subv = (const float4*)(sub + (size_t)b * ENC_);
    const float4 sv0 = subv[lane], sv1 = subv[lane + 32],
                 sv2 = subv[lane + 64], sv3 = subv[lane + 96];

    const float* encB = enc + (size_t)b * S_ * ENC_;

    float runmax = -__builtin_inff();
    float denom  = 0.0f;
    float4 a0 = {0,0,0,0}, a1 = {0,0,0,0}, a2 = {0,0,0,0}, a3 = {0,0,0,0};

    for (int i = 0; i < ROWS_PER_WAVE; ++i) {
        const int r = rowBase + w + WAVES * i;
        const float4* row = (const float4*)(encB + (size_t)r * ENC_);
        const float4 v0 = row[lane], v1 = row[lane + 32],
                     v2 = row[lane + 64], v3 = row[lane + 96];

        float part = v0.x*sv0.x + v0.y*sv0.y + v0.z*sv0.z + v0.w*sv0.w;
        part      += v1.x*sv1.x + v1.y*sv1.y + v1.z*sv1.z + v1.w*sv1.w;
        part      += v2.x*sv2.x + v2.y*sv2.y + v2.z*sv2.z + v2.w*sv2.w;
        part      += v3.x*sv3.x + v3.y*sv3.y + v3.z*sv3.z + v3.w*sv3.w;
#pragma unroll
        for (int m = 16; m > 0; m >>= 1) part += __shfl_xor(part, m, 32);
        const float score = part;
        if (lane == 0) attn_raw[(size_t)b * S_ + r] = score;   // raw; K4 normalizes

        const float nm = fmaxf(runmax, score);
        const float sc = __expf(runmax - nm);   // 0 on first iteration
        const float p  = __expf(score - nm);
        denom = denom * sc + p;
        a0.x = fmaf(p, v0.x, a0.x * sc); a0.y = fmaf(p, v0.y, a0.y * sc);
        a0.z = fmaf(p, v0.z, a0.z * sc); a0.w = fmaf(p, v0.w, a0.w * sc);
        a1.x = fmaf(p, v1.x, a1.x * sc); a1.y = fmaf(p, v1.y, a1.y * sc);
        a1.z = fmaf(p, v1.z, a1.z * sc); a1.w = fmaf(p, v1.w, a1.w * sc);
        a2.x = fmaf(p, v2.x, a2.x * sc); a2.y = fmaf(p, v2.y, a2.y * sc);
        a2.z = fmaf(p, v2.z, a2.z * sc); a2.w = fmaf(p, v2.w, a2.w * sc);
        a3.x = fmaf(p, v3.x, a3.x * sc); a3.y = fmaf(p, v3.y, a3.y * sc);
        a3.z = fmaf(p, v3.z, a3.z * sc); a3.w = fmaf(p, v3.w, a3.w * sc);
        runmax = nm;
    }

    ((float4*)&sacc[w][0])[lane]      = a0;
    ((float4*)&sacc[w][0])[lane + 32] = a1;
    ((float4*)&sacc[w][0])[lane + 64] = a2;
    ((float4*)&sacc[w][0])[lane + 96] = a3;
    if (lane == 0) { smax[w] = runmax; sden[w] = denom; }
    __syncthreads();

    // Combine the 8 wave partials into one record per block.
    float bmax = smax[0];
#pragma unroll
    for (int j = 1; j < WAVES; ++j) bmax = fmaxf(bmax, smax[j]);
    float scl[WAVES];
    float D = 0.0f;
#pragma unroll
    for (int j = 0; j < WAVES; ++j) { scl[j] = __expf(smax[j] - bmax); D += sden[j] * scl[j]; }

    float* rec = recs + (size_t)(b * CHUNKS + chunk) * REC_STRIDE;
    for (int e = tid; e < ENC_; e += 256) {
        float s = 0.0f;
#pragma unroll
        for (int j = 0; j < WAVES; ++j) s += sacc[j][e] * scl[j];
        rec[2 + e] = s;
    }
    if (tid == 0) { rec[0] = bmax; rec[1] = D; }
}

// ---------------------------------------------------------------------------
// Kernel 3: merge the 8 chunk records per batch -> sumResult + (max, 1/denom)
// ---------------------------------------------------------------------------
__global__ __launch_bounds__(256) void attn_reduce(const float* __restrict__ recs,
                                                   float* __restrict__ out_sum,
                                                   float* __restrict__ stats) {
    const int b   = blockIdx.x;
    const int tid = threadIdx.x;
    const float* base = recs + (size_t)b * CHUNKS * REC_STRIDE;

    float M = base[0];
#pragma unroll
    for (int c = 1; c < CHUNKS; ++c) M = fmaxf(M, base[(size_t)c * REC_STRIDE]);
    float scl[CHUNKS];
    float D = 0.0f;
#pragma unroll
    for (int c = 0; c < CHUNKS; ++c) {
        scl[c] = __expf(base[(size_t)c * REC_STRIDE] - M);
        D += base[(size_t)c * REC_STRIDE + 1] * scl[c];
    }
    const float invD = 1.0f / D;

    for (int e = tid; e < ENC_; e += 256) {
        float s = 0.0f;
#pragma unroll
        for (int c = 0; c < CHUNKS; ++c) s += base[(size_t)c * REC_STRIDE + 2 + e] * scl[c];
        out_sum[(size_t)b * ENC_ + e] = s * invD;
    }
    if (tid == 0) { stats[2 * b] = M; stats[2 * b + 1] = invD; }
}

// ---------------------------------------------------------------------------
// Kernel 4: normalize raw scores in place -> final attn probabilities.
// ---------------------------------------------------------------------------
__global__ __launch_bounds__(256) void attn_norm(float* __restrict__ attn,
                                                 const float* __restrict__ stats) {
    const int idx = blockIdx.x * 256 + threadIdx.x;   // 0 .. B*S-1
    const int b   = idx >> 12;                        // S = 4096
    const float M    = stats[2 * b];
    const float invD = stats[2 * b + 1];
    attn[idx] = __expf(attn[idx] - M) * invD;
}

// ---------------------------------------------------------------------------
extern "C" void kernel_launch(void* const* d_in, const int* in_sizes, int n_in,
                              void* d_out, int out_size, void* d_ws, size_t ws_size,
                              hipStream_t stream) {
    const float* dec = (const float*)d_in[0];   // [B,1,DEC]
    const float* enc = (const float*)d_in[1];   // [B,S,ENC]
    const float* W   = (const float*)d_in[2];   // [ENC,DEC]

    float* attn = (float*)d_out;                         // [B,1,S]  = 262144
    float* sumr = (float*)d_out + (size_t)B_ * S_;       // [B,ENC]  = 32768

    float* ws    = (float*)d_ws;
    float* sub   = ws;                                           // B*ENC = 32768 floats
    float* recs  = sub + (size_t)B_ * ENC_;                      // B*CHUNKS*520 floats
    float* stats = recs + (size_t)B_ * CHUNKS * REC_STRIDE;      // 2*B floats

    sub_gemm_wmma<<<dim3(B_ / 16, ENC_ / 16), 32, 0, stream>>>(dec, W, sub);
    attn_pass1  <<<dim3(B_, CHUNKS), 256, 0, stream>>>(enc, sub, attn, recs);
    attn_reduce <<<B_, 256, 0, stream>>>(recs, sumr, stats);
    attn_norm   <<<(B_ * S_) / 256, 256, 0, stream>>>(attn, stats);
}